// VAELoss_33320356283101
// MI455X (gfx1250) — compile-verified
//
#include <hip/hip_runtime.h>

typedef float v2f __attribute__((ext_vector_type(2)));
typedef float v8f __attribute__((ext_vector_type(8)));
typedef float f4  __attribute__((ext_vector_type(4)));

#define IMG_ELEMS 196608     // C*H*W = 3*256*256
#define IMG_F4    (IMG_ELEMS / 4)               // 49152 float4 per image
#define CHUNKS    16                            // blocks per image
#define BLK       256                           // 8 wave32 per block
#define F4_PER_THREAD (IMG_F4 / CHUNKS / BLK)   // 12, exact (no tail)
#define NBATCH    256
#define DLAT      512

#define LN2F        0.69314718055994530942f
#define NEG_LN2F   (-0.69314718055994530942f)
// clamp(-100) mapped into log2 domain: -100 / ln2
#define LOG2_CLAMP (-144.26950408889634f)

// ---------------------------------------------------------------------------
// Kernel 1: per-(image, chunk) BCE partial sums, computed in the log2 domain.
//   bce = -( g*ln(p) + (1-g)*ln(1-p) )  with ln terms clamped at -100
//       = -ln2 * ( g*(L2p - L2q) + L2q ),  L2* clamped at -100/ln2
// Raw v_log_f32 (log2) per element; positive accumulation through
// V_WMMA_F32_16X16X4_F32 against a ones B-matrix; single -ln2 scale at the
// epilogue. Streams preds/gts with NT b128 loads (no reuse; spare the L2).
// ---------------------------------------------------------------------------
__global__ __launch_bounds__(BLK) void bce_partial_kernel(
    const float* __restrict__ preds, const float* __restrict__ gts,
    float* __restrict__ partials)
{
    const int img   = blockIdx.y;
    const int chunk = blockIdx.x;
    const int t     = threadIdx.x;

    const long base4 = (long)img * IMG_F4 + (long)chunk * (IMG_F4 / CHUNKS);
    const f4* __restrict__ p4 = (const f4*)preds + base4 + t;
    const f4* __restrict__ g4 = (const f4*)gts   + base4 + t;

    const v2f ones = {1.0f, 1.0f};
    v8f c0 = {};   // accumulates A = (v0,v1) fragments
    v8f c1 = {};   // accumulates A = (v2,v3) fragments

#pragma unroll
    for (int k = 0; k < F4_PER_THREAD; ++k) {
        f4 p = __builtin_nontemporal_load(p4 + (long)k * BLK);
        f4 g = __builtin_nontemporal_load(g4 + (long)k * BLK);

        float v[4];
#pragma unroll
        for (int j = 0; j < 4; ++j) {
            float pp  = p[j];
            float gg  = g[j];
            // raw hardware log2 (v_log_f32); inputs are safely in-range
            float l2p = fmaxf(__builtin_amdgcn_logf(pp),        LOG2_CLAMP);
            float l2q = fmaxf(__builtin_amdgcn_logf(1.0f - pp), LOG2_CLAMP);
            // positive log2-domain accumulation; -ln2 applied once at the end
            v[j] = fmaf(gg, l2p - l2q, l2q);
        }

        v2f a01 = {v[0], v[1]};
        v2f a23 = {v[2], v[3]};
        // D = A x Ones + C : every column of C accumulates the wave's sums.
        c0 = __builtin_amdgcn_wmma_f32_16x16x4_f32(
                 false, a01, false, ones, (short)0, c0, false, false);
        c1 = __builtin_amdgcn_wmma_f32_16x16x4_f32(
                 false, a23, false, ones, (short)0, c1, false, false);
    }

    // Epilogue: lane n (0-15) holds C[0..7][n], lane n+16 holds C[8..15][n].
    // Per-lane sum of 8 VGPRs + xor-16 shuffle == full wave sum in every lane.
    float s = 0.0f;
#pragma unroll
    for (int i = 0; i < 8; ++i) s += c0[i] + c1[i];
    s += __shfl_xor(s, 16, 32);

    __shared__ float wsum[BLK / 32];
    if ((t & 31) == 0) wsum[t >> 5] = s;
    __syncthreads();
    if (t == 0) {
        float tot = 0.0f;
#pragma unroll
        for (int w = 0; w < BLK / 32; ++w) tot += wsum[w];
        partials[img * CHUNKS + chunk] = tot * NEG_LN2F;  // log2 -> -ln sum
    }
}

// ---------------------------------------------------------------------------
// Kernel 2: per-image finalize. Reduces the 16 BCE partials, computes the
// diagonal-Gaussian KL over D=512 (log2 domain, scaled once), writes
// rec/kl/total in reference output order.
// ---------------------------------------------------------------------------
__global__ __launch_bounds__(128) void finalize_kernel(
    const float* __restrict__ mean, const float* __restrict__ var,
    const float* __restrict__ partials, float* __restrict__ out)
{
    const int b    = blockIdx.x;
    const int t    = threadIdx.x;
    const int lane = t & 31;
    const int wid  = t >> 5;

    // KL accumulation: 512 elems over 128 threads (coalesced, stride 128)
    float asq = 0.0f, alg2 = 0.0f;
#pragma unroll
    for (int k = 0; k < DLAT / 128; ++k) {
        int j   = t + k * 128;
        float m = mean[b * DLAT + j];
        float s = var [b * DLAT + j] + 1e-6f;
        asq  = fmaf(s, s, asq);
        asq  = fmaf(m, m, asq);
        alg2 += __builtin_amdgcn_logf(s);   // log2; s in [0.05, 1.5]
    }
#pragma unroll
    for (int off = 16; off > 0; off >>= 1) {
        asq  += __shfl_xor(asq,  off, 32);
        alg2 += __shfl_xor(alg2, off, 32);
    }
    __shared__ float rsq[4], rlg[4];
    if (lane == 0) { rsq[wid] = asq; rlg[wid] = alg2; }

    // BCE partial reduction inside wave 0 (lanes >= CHUNKS contribute 0)
    float pr = (t < CHUNKS) ? partials[b * CHUNKS + t] : 0.0f;
    if (wid == 0) {
#pragma unroll
        for (int off = 16; off > 0; off >>= 1) pr += __shfl_xor(pr, off, 32);
    }
    __syncthreads();

    if (t == 0) {
        float sq  = rsq[0] + rsq[1] + rsq[2] + rsq[3];
        float lg  = (rlg[0] + rlg[1] + rlg[2] + rlg[3]) * LN2F;  // log2 -> ln
        float kl  = 0.5f * (sq - (float)DLAT) - lg;   // 0.5*(Σ(s²+m²) - D - 2Σln s)
        float rec = pr * (1.0f / (float)IMG_ELEMS);
        out[b]              = rec;                    // rec_loss
        out[NBATCH + b]     = kl;                     // kl_loss
        out[2 * NBATCH + b] = fmaf(500.0f, rec, kl);  // total loss
    }
}

extern "C" void kernel_launch(void* const* d_in, const int* in_sizes, int n_in,
                              void* d_out, int out_size, void* d_ws, size_t ws_size,
                              hipStream_t stream)
{
    const float* preds = (const float*)d_in[0];
    const float* gts   = (const float*)d_in[1];
    const float* mean  = (const float*)d_in[2];
    const float* var   = (const float*)d_in[3];
    float* out      = (float*)d_out;
    float* partials = (float*)d_ws;   // NBATCH*CHUNKS floats = 16 KB

    dim3 grid1(CHUNKS, NBATCH);
    bce_partial_kernel<<<grid1, BLK, 0, stream>>>(preds, gts, partials);
    finalize_kernel<<<NBATCH, 128, 0, stream>>>(mean, var, partials, out);
}